// SPDNetClassic_9818295239295
// MI455X (gfx1250) — compile-verified
//
#include <hip/hip_runtime.h>
#include <math.h>

// Problem constants (match reference)
#define B_   8192
#define C_   21
#define T_   256
#define D1_  10
#define D2_  5
#define D3_  2
#define XS   260      // LDS row stride (floats): 16B-aligned rows, bank-stride 4
#define SS   33       // padded stride for the 32x32 covariance tile
#define REEIG_EPS 1e-4f
#define LOG_EPS   1e-8f

typedef float v2f __attribute__((ext_vector_type(2)));
typedef float v8f __attribute__((ext_vector_type(8)));

// --- LDS byte offset of a __shared__ object (generic -> addrspace(3) -> int)
__device__ __forceinline__ unsigned lds_off_bytes(const void* p) {
    return (unsigned)(unsigned long long)
        ((const __attribute__((address_space(3))) char*)p);
}

// --- CDNA5 async global->LDS copy, 16B per lane (GVS mode: saddr + vaddr) ---
__device__ __forceinline__ void async_ld_b128(unsigned lds_off,
                                              const void* sbase,
                                              unsigned vbyteoff) {
    asm volatile("global_load_async_to_lds_b128 %0, %1, %2"
                 :: "v"(lds_off), "v"(vbyteoff), "s"(sbase)
                 : "memory");
}
__device__ __forceinline__ void wait_async0() {
    asm volatile("s_wait_asynccnt 0" ::: "memory");
}

// --- BiMap: M(no x no) = W^T * M(ni x ni) * W,  W is (ni x no) row-major ---
__device__ void bimap(float* M, float* Tb, const float* __restrict__ W,
                      int ni, int no, int lane) {
    for (int e = lane; e < ni * no; e += 32) {        // Tb = M * W
        int i = e / no, j = e % no;
        float s = 0.f;
        for (int k = 0; k < ni; ++k) s += M[i * ni + k] * W[k * no + j];
        Tb[e] = s;
    }
    __syncthreads();
    for (int e = lane; e < no * no; e += 32) {        // M = W^T * Tb
        int i = e / no, j = e % no;
        float s = 0.f;
        for (int k = 0; k < ni; ++k) s += W[k * no + i] * Tb[k * no + j];
        M[e] = s;
    }
    __syncthreads();
}

// --- lane-parallel cyclic Jacobi eigh on LDS matrix A (n x n), n <= 10 ---
__device__ void jacobi(float* A, float* V, float* w, int n, int sweeps, int lane) {
    for (int e = lane; e < n * n; e += 32) V[e] = ((e % (n + 1)) == 0) ? 1.f : 0.f;
    __syncthreads();
    for (int s = 0; s < sweeps; ++s)
        for (int p = 0; p < n - 1; ++p)
            for (int q = p + 1; q < n; ++q) {
                float apq = A[p * n + q];             // uniform across lanes
                float cth = 1.f, sth = 0.f;
                if (fabsf(apq) > 1e-20f) {
                    float tau = (A[q * n + q] - A[p * n + p]) / (2.f * apq);
                    float t = (tau >= 0.f ? 1.f : -1.f) /
                              (fabsf(tau) + sqrtf(1.f + tau * tau));
                    cth = rsqrtf(1.f + t * t);
                    sth = t * cth;
                }
                __syncthreads();
                if (lane < n) {                        // column rotation
                    float aip = A[lane * n + p], aiq = A[lane * n + q];
                    A[lane * n + p] = cth * aip - sth * aiq;
                    A[lane * n + q] = sth * aip + cth * aiq;
                }
                __syncthreads();
                if (lane < n) {                        // row rotation
                    float api = A[p * n + lane], aqi = A[q * n + lane];
                    A[p * n + lane] = cth * api - sth * aqi;
                    A[q * n + lane] = sth * api + cth * aqi;
                }
                __syncthreads();
                if (lane < n) {                        // accumulate V
                    float vip = V[lane * n + p], viq = V[lane * n + q];
                    V[lane * n + p] = cth * vip - sth * viq;
                    V[lane * n + q] = sth * vip + cth * viq;
                }
                __syncthreads();
            }
    if (lane < n) w[lane] = A[lane * n + lane];
    __syncthreads();
}

// --- A = V * diag(f(w)) * V^T, f = clamp (ReEig) or log(clamp) (LogEig) ---
__device__ void recompose(float* A, const float* V, const float* w, int n,
                          bool logmap, float eps, int lane) {
    for (int e = lane; e < n * n; e += 32) {
        int i = e / n, j = e % n;
        float s = 0.f;
        for (int k = 0; k < n; ++k) {
            float wk = fmaxf(w[k], eps);
            if (logmap) wk = logf(wk);
            s += V[i * n + k] * wk * V[j * n + k];
        }
        A[e] = s;
    }
    __syncthreads();
}

__global__ __launch_bounds__(32)
void spdnet_kernel(const float* __restrict__ x, const float* __restrict__ alpha,
                   const float* __restrict__ W1, const float* __restrict__ W2,
                   const float* __restrict__ W3, const float* __restrict__ Wc,
                   const float* __restrict__ bc, float* __restrict__ out) {
    __shared__ float xs[32 * XS];      // padded 32 x 256 centered tile
    __shared__ float S [32 * SS];      // 32 x 32 covariance (WMMA output)
    __shared__ float Ma[C_ * C_];      // current SPD matrix (max 21x21)
    __shared__ float Tb[C_ * D1_];     // bimap temp (max 21x10)
    __shared__ float Va[D1_ * D1_];    // eigenvectors (max 10x10)
    __shared__ float wv[D1_];          // eigenvalues

    const int lane = threadIdx.x;
    const int b = blockIdx.x;
    const float* xb = x + (size_t)b * (C_ * T_);

    // ---- Phase 1: async HBM -> LDS (21x256 f32 = 1344 x 16B) ----
    for (int v = lane; v < (C_ * T_) / 4; v += 32) {
        int f = v * 4;
        int row = f >> 8;              // f / 256
        int col = f & 255;
        async_ld_b128(lds_off_bytes(&xs[row * XS + col]), xb,
                      (unsigned)(f * 4));
    }
    wait_async0();
    __syncthreads();

    // zero-pad rows 21..31 for the WMMA tiles
    for (int r = C_; r < 32; ++r)
        for (int k = lane; k < T_; k += 32) xs[r * XS + k] = 0.f;
    __syncthreads();

    // ---- Phase 2: mean-center rows (lane r handles row r) ----
    if (lane < C_) {
        float s = 0.f;
        for (int k = 0; k < T_; ++k) s += xs[lane * XS + k];
        float m = s * (1.f / T_);
        for (int k = 0; k < T_; ++k) xs[lane * XS + k] -= m;
    }
    __syncthreads();

    // ---- Phase 3: S = xc * xc^T / (T-1) via V_WMMA_F32_16X16X4_F32 ----
    // A frag (16x4 f32): lane L holds row M=L&15, K = 2*(L/16) + {0,1}
    // B frag (4x16 f32): lane L holds col N=L&15,  K = 2*(L/16) + {0,1}
    // C/D (16x16 f32):   VGPR r -> row r (lanes 0-15) / r+8 (lanes 16-31)
    const float invTm1 = 1.f / (float)(T_ - 1);
    const int mm = lane & 15;
    const int kb = (lane >> 4) << 1;
    for (int ti = 0; ti < 2; ++ti)
        for (int tj = 0; tj < 2; ++tj) {
            v8f acc = {0.f, 0.f, 0.f, 0.f, 0.f, 0.f, 0.f, 0.f};
            const float* ra = &xs[(ti * 16 + mm) * XS];
            const float* rb = &xs[(tj * 16 + mm) * XS];
            for (int k0 = 0; k0 < T_; k0 += 4) {
                v2f a, bv;
                a.x  = ra[k0 + kb]; a.y  = ra[k0 + kb + 1];
                bv.x = rb[k0 + kb]; bv.y = rb[k0 + kb + 1];
                acc = __builtin_amdgcn_wmma_f32_16x16x4_f32(
                    false, a, false, bv, (short)0, acc, false, false);
            }
            int rbase = ti * 16 + ((lane >> 4) << 3);
            int col   = tj * 16 + mm;
            for (int r = 0; r < 8; ++r)
                S[(rbase + r) * SS + col] = acc[r] * invTm1;
        }
    __syncthreads();

    // ---- Phase 4: shrinkage S' = (1-a)S + a*mu*I into Ma (21x21) ----
    if (lane == 0) {
        float tr = 0.f;
        for (int i = 0; i < C_; ++i) tr += S[i * SS + i];
        wv[0] = tr / (float)C_;
    }
    __syncthreads();
    {
        const float a = alpha[0];
        const float mu = wv[0];
        for (int e = lane; e < C_ * C_; e += 32) {
            int i = e / C_, j = e % C_;
            Ma[e] = (1.f - a) * S[i * SS + j] + ((i == j) ? a * mu : 0.f);
        }
    }
    __syncthreads();

    // ---- Phase 5: BiMap+ReEig x2, then BiMap+LogEig ----
    bimap(Ma, Tb, W1, C_,  D1_, lane);
    jacobi(Ma, Va, wv, D1_, 8, lane);
    recompose(Ma, Va, wv, D1_, false, REEIG_EPS, lane);

    bimap(Ma, Tb, W2, D1_, D2_, lane);
    jacobi(Ma, Va, wv, D2_, 7, lane);
    recompose(Ma, Va, wv, D2_, false, REEIG_EPS, lane);

    bimap(Ma, Tb, W3, D2_, D3_, lane);
    jacobi(Ma, Va, wv, D3_, 4, lane);
    recompose(Ma, Va, wv, D3_, true, LOG_EPS, lane);

    // ---- Phase 6: feat = upper-tri(L) = [L00, L01, L11]; out = feat@Wc^T+bc
    if (lane < 2) {
        float f0 = Ma[0], f1 = Ma[1], f2 = Ma[3];
        out[(size_t)b * 2 + lane] =
            Wc[lane * 3 + 0] * f0 + Wc[lane * 3 + 1] * f1 +
            Wc[lane * 3 + 2] * f2 + bc[lane];
    }
}

extern "C" void kernel_launch(void* const* d_in, const int* in_sizes, int n_in,
                              void* d_out, int out_size, void* d_ws, size_t ws_size,
                              hipStream_t stream) {
    const float* x     = (const float*)d_in[0];
    const float* alpha = (const float*)d_in[1];
    const float* W1    = (const float*)d_in[2];
    const float* W2    = (const float*)d_in[3];
    const float* W3    = (const float*)d_in[4];
    const float* Wc    = (const float*)d_in[5];
    const float* bc    = (const float*)d_in[6];
    (void)in_sizes; (void)n_in; (void)out_size; (void)d_ws; (void)ws_size;
    spdnet_kernel<<<B_, 32, 0, stream>>>(x, alpha, W1, W2, W3, Wc, bc,
                                         (float*)d_out);
}